// PyramidROIAlign_43104291782861
// MI455X (gfx1250) — compile-verified
//
#include <hip/hip_runtime.h>
#include <math.h>

#define PH 7
#define CCH 256   // channels

// ---- CDNA5 async global->LDS path detection (device pass only) ----
#if defined(__gfx1250__) && __has_builtin(__builtin_amdgcn_global_load_async_to_lds_b128) && __has_builtin(__builtin_amdgcn_s_wait_asynccnt)
#define USE_ASYNC 1
#else
#define USE_ASYNC 0
#endif

#if USE_ASYNC
// Matches the builtin's parameter type from the clang diagnostic:
//   '__attribute__((__vector_size__(4 * sizeof(int)))) int __device__ *'
typedef int v4i __attribute__((vector_size(4 * sizeof(int))));
typedef __attribute__((address_space(1))) v4i* gv4p;   // global (prints as __device__)
typedef __attribute__((address_space(3))) v4i* lv4p;   // LDS    (prints as __shared__)
#endif

__device__ __forceinline__ float4 lerp4(float4 a, float4 b, float t) {
    float4 r;
    r.x = a.x + (b.x - a.x) * t;
    r.y = a.y + (b.y - a.y) * t;
    r.z = a.z + (b.z - a.z) * t;
    r.w = a.w + (b.w - a.w) * t;
    return r;
}

__global__ __launch_bounds__(256) void pyramid_roi_align_kernel(
    const float* __restrict__ boxes,  // [N,4] (y1,x1,y2,x2) normalized
    const float* __restrict__ meta,   // [93], meta[4]=imgH, meta[5]=imgW
    const float* __restrict__ P2, const float* __restrict__ P3,
    const float* __restrict__ P4, const float* __restrict__ P5,
    int H2, int H3, int H4, int H5,
    float* __restrict__ out,          // [N,7,7,256]
    int nroi)
{
    const int roi = blockIdx.x;
    if (roi >= nroi) return;
    const int tid = threadIdx.x;
    const int s = tid >> 6;   // subgroup 0..3 (one 7x7 position at a time)
    const int l = tid & 63;   // lane within subgroup: float4 over 256 channels

    const float by1 = boxes[roi * 4 + 0];
    const float bx1 = boxes[roi * 4 + 1];
    const float by2 = boxes[roi * 4 + 2];
    const float bx2 = boxes[roi * 4 + 3];
    const float bh = by2 - by1;
    const float bw = bx2 - bx1;

    // roi_level = clip(4 + round(log2(sqrt(h*w) / (224/sqrt(area)))), 2, 5)
    const float area = meta[4] * meta[5];
    const float lvlf = log2f(sqrtf(bh * bw) / (224.0f / sqrtf(area)));
    int level = 4 + (int)rintf(lvlf);   // rintf == round-half-even == jnp.round
    level = level < 2 ? 2 : (level > 5 ? 5 : level);

    const float* fmap = P2; int H = H2;
    if (level == 3) { fmap = P3; H = H3; }
    if (level == 4) { fmap = P4; H = H4; }
    if (level == 5) { fmap = P5; H = H5; }
    const int W = H;                 // square maps
    const float fHm1 = (float)(H - 1);

    // per-position sampling geometry (matches reference crop_and_resize)
    auto corner = [&](int p, int& y0, int& x0, int& y1i, int& x1i, float& wy, float& wx) {
        const int py = p / PH, px = p - py * PH;
        const float ys = (by1 + ((float)py / 6.0f) * bh) * fHm1;
        const float xs = (bx1 + ((float)px / 6.0f) * bw) * fHm1;
        float y0f = floorf(ys); y0f = y0f < 0.0f ? 0.0f : (y0f > fHm1 ? fHm1 : y0f);
        float x0f = floorf(xs); x0f = x0f < 0.0f ? 0.0f : (x0f > fHm1 ? fHm1 : x0f);
        y0 = (int)y0f; x0 = (int)x0f;
        y1i = (y0 + 1 < H) ? y0 + 1 : H - 1;
        x1i = (x0 + 1 < W) ? x0 + 1 : W - 1;
        wy = ys - y0f; wx = xs - x0f;
    };

    auto blend_store = [&](int p, float4 v00, float4 v01, float4 v10, float4 v11,
                           float wy, float wx) {
        const float4 top = lerp4(v00, v01, wx);
        const float4 bot = lerp4(v10, v11, wx);
        const float4 res = lerp4(top, bot, wy);
        float4* op = (float4*)(out + ((size_t)roi * (PH * PH) + p) * CCH) + l;
        *op = res;
    };

#if USE_ASYNC
    // Double-buffered async gather: 4 corner rows (1KB each) -> LDS, ASYNCcnt-tracked.
    __shared__ float4 tile[2][4][4][64];   // [buf][subgroup][corner][lane] = 32 KB

    auto issue = [&](int p, int buf) {
        int y0, x0, y1i, x1i; float wy, wx;
        corner(p, y0, x0, y1i, x1i, wy, wx);
        const float4* c00 = (const float4*)(fmap + ((size_t)y0  * W + x0 ) * CCH) + l;
        const float4* c01 = (const float4*)(fmap + ((size_t)y0  * W + x1i) * CCH) + l;
        const float4* c10 = (const float4*)(fmap + ((size_t)y1i * W + x0 ) * CCH) + l;
        const float4* c11 = (const float4*)(fmap + ((size_t)y1i * W + x1i) * CCH) + l;
        __builtin_amdgcn_global_load_async_to_lds_b128((gv4p)c00, (lv4p)&tile[buf][s][0][l], 0, 0);
        __builtin_amdgcn_global_load_async_to_lds_b128((gv4p)c01, (lv4p)&tile[buf][s][1][l], 0, 0);
        __builtin_amdgcn_global_load_async_to_lds_b128((gv4p)c10, (lv4p)&tile[buf][s][2][l], 0, 0);
        __builtin_amdgcn_global_load_async_to_lds_b128((gv4p)c11, (lv4p)&tile[buf][s][3][l], 0, 0);
    };

    auto emit = [&](int p, int buf) {
        int y0, x0, y1i, x1i; float wy, wx;
        corner(p, y0, x0, y1i, x1i, wy, wx);
        const float4 v00 = tile[buf][s][0][l];
        const float4 v01 = tile[buf][s][1][l];
        const float4 v10 = tile[buf][s][2][l];
        const float4 v11 = tile[buf][s][3][l];
        blend_store(p, v00, v01, v10, v11, wy, wx);
    };

    int p = s, buf = 0;
    issue(p, buf);                               // prologue: 4 gathers in flight
    while (p + 4 < PH * PH) {                    // wave-uniform loop (s uniform per wave)
        issue(p + 4, buf ^ 1);                   // 8 in flight
        __builtin_amdgcn_s_wait_asynccnt(4);     // oldest 4 (current buf) landed in LDS
        asm volatile("" ::: "memory");
        emit(p, buf);
        buf ^= 1; p += 4;
    }
    __builtin_amdgcn_s_wait_asynccnt(0);
    asm volatile("" ::: "memory");
    emit(p, buf);
#else
    // Fallback: direct coalesced float4 gathers (still correct, no async path).
    for (int p = s; p < PH * PH; p += 4) {
        int y0, x0, y1i, x1i; float wy, wx;
        corner(p, y0, x0, y1i, x1i, wy, wx);
        const float4 v00 = *((const float4*)(fmap + ((size_t)y0  * W + x0 ) * CCH) + l);
        const float4 v01 = *((const float4*)(fmap + ((size_t)y0  * W + x1i) * CCH) + l);
        const float4 v10 = *((const float4*)(fmap + ((size_t)y1i * W + x0 ) * CCH) + l);
        const float4 v11 = *((const float4*)(fmap + ((size_t)y1i * W + x1i) * CCH) + l);
        blend_store(p, v00, v01, v10, v11, wy, wx);
    }
#endif
}

static inline int side_of(int flat_elems) {
    // feature maps are [1,H,H,256]; recover H
    double hw = (double)flat_elems / (double)CCH;
    return (int)(sqrt(hw) + 0.5);
}

extern "C" void kernel_launch(void* const* d_in, const int* in_sizes, int n_in,
                              void* d_out, int out_size, void* d_ws, size_t ws_size,
                              hipStream_t stream) {
    const float* boxes = (const float*)d_in[0];  // ROIboxes [1,N,4]
    const float* meta  = (const float*)d_in[1];  // image_meta [1,93]
    const float* P2    = (const float*)d_in[2];
    const float* P3    = (const float*)d_in[3];
    const float* P4    = (const float*)d_in[4];
    const float* P5    = (const float*)d_in[5];

    const int nroi = in_sizes[0] / 4;
    const int H2 = side_of(in_sizes[2]);
    const int H3 = side_of(in_sizes[3]);
    const int H4 = side_of(in_sizes[4]);
    const int H5 = side_of(in_sizes[5]);

    pyramid_roi_align_kernel<<<nroi, 256, 0, stream>>>(
        boxes, meta, P2, P3, P4, P5, H2, H3, H4, H5, (float*)d_out, nroi);
}